// SGC_64561948393904
// MI455X (gfx1250) — compile-verified
//
#include <hip/hip_runtime.h>

typedef __attribute__((ext_vector_type(2))) float v2f;
typedef __attribute__((ext_vector_type(8))) float v8f;

#define N_NODES   100000
#define N_EDGES   1600000
#define D_FEAT    128
#define N_CLASSES 2

// ---------- degree / normalization ----------

__global__ void k_set_deg(float* __restrict__ deg) {
    int n = blockIdx.x * blockDim.x + threadIdx.x;
    if (n < N_NODES) deg[n] = 1.0f;          // self-loop
}

__global__ void k_deg_accum(const long long* __restrict__ col, float* __restrict__ deg) {
    int e = blockIdx.x * blockDim.x + threadIdx.x;
    if (e < N_EDGES) {
        int t = (int)col[e];
        unsafeAtomicAdd(&deg[t], 1.0f);      // hardware global_atomic_add_f32
    }
}

__global__ void k_dinv(float* __restrict__ deg) {
    int n = blockIdx.x * blockDim.x + threadIdx.x;
    if (n < N_NODES) deg[n] = rsqrtf(deg[n]); // deg >= 1 always (self-loops)
}

// ---------- one propagation hop ----------

// dst[n][:] = dinv[n]^2 * src[n][:]   (self-loop contribution), float4 per thread
__global__ void k_hop_init(const float* __restrict__ src, const float* __restrict__ dinv,
                           float* __restrict__ dst) {
    int i = blockIdx.x * blockDim.x + threadIdx.x;   // one float4
    const int total = N_NODES * (D_FEAT / 4);
    if (i < total) {
        int n = i >> 5;                               // 32 float4 per node
        float w = dinv[n]; w *= w;
        float4 v = ((const float4*)src)[i];
        v.x *= w; v.y *= w; v.z *= w; v.w *= w;
        ((float4*)dst)[i] = v;
    }
}

// Edge scatter: each wave grabs a batch of 32 edges (lane e = base+lane loads
// metadata coalesced, computes w), then iterates the 32 edges broadcasting
// (s, t, w) via wave shuffles. Per edge: 1x global_load_b128 per lane +
// 4x global_atomic_add_f32 per lane. N_EDGES % 32 == 0 -> no tail handling.
__global__ void k_hop_scatter(const long long* __restrict__ row,
                              const long long* __restrict__ col,
                              const float* __restrict__ dinv,
                              const float* __restrict__ src,
                              float* __restrict__ dst) {
    int lane   = threadIdx.x & 31;
    int wave   = (blockIdx.x * blockDim.x + threadIdx.x) >> 5;
    int nwaves = (gridDim.x * blockDim.x) >> 5;

    for (int base = wave * 32; base < N_EDGES; base += nwaves * 32) {
        int e = base + lane;                 // always < N_EDGES (E % 32 == 0)
        int s = (int)row[e];
        int t = (int)col[e];
        float w = dinv[s] * dinv[t];

        #pragma unroll 4
        for (int j = 0; j < 32; ++j) {
            int   sj = __shfl(s, j, 32);
            int   tj = __shfl(t, j, 32);
            float wj = __shfl(w, j, 32);
            const float4 v = ((const float4*)(src + (size_t)sj * D_FEAT))[lane];
            float* o = dst + (size_t)tj * D_FEAT + lane * 4;
            unsafeAtomicAdd(o + 0, wj * v.x);
            unsafeAtomicAdd(o + 1, wj * v.y);
            unsafeAtomicAdd(o + 2, wj * v.z);
            unsafeAtomicAdd(o + 3, wj * v.w);
        }
    }
}

// ---------- WMMA f32 classifier + log-softmax ----------
// Each wave: 16-node tile, D = X(16x128) * Wpad^T(128x16, rows 2..15 zero) via
// 32x V_WMMA_F32_16X16X4_F32 (full f32 precision). B comes from a zero-padded
// LDS tile so the inner loop is branch-free (plain ds_load_b64, no exec masks).
__global__ void k_classify(const float* __restrict__ x,
                           const float* __restrict__ W,
                           const float* __restrict__ bias,
                           float* __restrict__ out) {
    __shared__ float Wpad[16][D_FEAT];     // rows 0..1 = W, rows 2..15 = 0
    __shared__ float tileLds[8][16][16];

    int tid = threadIdx.x;
    float* wflat = &Wpad[0][0];
    #pragma unroll
    for (int i = tid; i < 16 * D_FEAT; i += 256) wflat[i] = 0.0f;
    __syncthreads();
    if (tid < N_CLASSES * D_FEAT) wflat[tid] = W[tid];   // rows 0,1
    __syncthreads();

    int wave = tid >> 5;
    int lane = tid & 31;
    int tile = blockIdx.x * 8 + wave;            // 16 nodes per tile
    bool valid = (tile * 16) < N_NODES;          // wave-uniform -> EXEC all ones inside

    if (valid) {
        int n16   = lane & 15;                   // A: row M / B: col N
        int khalf = (lane >> 4) << 1;            // lanes 0-15 -> K+0,1 ; 16-31 -> K+2,3
        const float* xrow = x + (size_t)(tile * 16 + n16) * D_FEAT;
        const float* wrow = &Wpad[n16][0];

        v8f acc = {};
        #pragma unroll
        for (int kb = 0; kb < D_FEAT; kb += 4) {
            v2f a = *(const v2f*)(xrow + kb + khalf);   // global_load_b64
            v2f b = *(const v2f*)(wrow + kb + khalf);   // ds_load_b64 (unpredicated)
            acc = __builtin_amdgcn_wmma_f32_16x16x4_f32(
                false, a, false, b, (short)0, acc, false, false);
        }
        // D layout: VGPR r -> M = r + (lane<16 ? 0 : 8), N = lane&15
        int mbase = (lane >> 4) << 3;
        #pragma unroll
        for (int r = 0; r < 8; ++r)
            tileLds[wave][mbase + r][n16] = acc[r];
    }
    __syncthreads();

    if (valid && lane < 16) {
        int node = tile * 16 + lane;
        float l0 = tileLds[wave][lane][0] + bias[0];
        float l1 = tileLds[wave][lane][1] + bias[1];
        float m  = fmaxf(l0, l1);
        float lse = m + logf(expf(l0 - m) + expf(l1 - m));
        out[node * 2 + 0] = l0 - lse;
        out[node * 2 + 1] = l1 - lse;
    }
}

// ---------- host ----------

extern "C" void kernel_launch(void* const* d_in, const int* in_sizes, int n_in,
                              void* d_out, int out_size, void* d_ws, size_t ws_size,
                              hipStream_t stream) {
    const float*     x    = (const float*)d_in[0];
    const long long* ei   = (const long long*)d_in[1];   // int64 [2, E]
    const float*     W    = (const float*)d_in[2];       // [2,128]
    const float*     bias = (const float*)d_in[3];       // [2]
    float*           out  = (float*)d_out;               // [N,2]

    const long long* row = ei;                 // sources
    const long long* col = ei + N_EDGES;       // targets

    // workspace: dinv[N] | buf1[N*128] | buf2[N*128]  (~103 MB)
    float* dinv = (float*)d_ws;
    float* buf1 = dinv + N_NODES;
    float* buf2 = buf1 + (size_t)N_NODES * D_FEAT;

    const int B = 256;
    k_set_deg  <<<(N_NODES + B - 1) / B, B, 0, stream>>>(dinv);
    k_deg_accum<<<(N_EDGES + B - 1) / B, B, 0, stream>>>(col, dinv);
    k_dinv     <<<(N_NODES + B - 1) / B, B, 0, stream>>>(dinv);

    const int initBlocks    = (N_NODES * (D_FEAT / 4) + B - 1) / B;
    const int scatterBlocks = 6250;   // 50000 waves x 32-edge batches = 1.6M edges

    // hop 1: x -> buf1
    k_hop_init   <<<initBlocks, B, 0, stream>>>(x, dinv, buf1);
    k_hop_scatter<<<scatterBlocks, B, 0, stream>>>(row, col, dinv, x, buf1);
    // hop 2: buf1 -> buf2
    k_hop_init   <<<initBlocks, B, 0, stream>>>(buf1, dinv, buf2);
    k_hop_scatter<<<scatterBlocks, B, 0, stream>>>(row, col, dinv, buf1, buf2);

    // classifier: 6250 tiles of 16 nodes, 8 waves/block
    const int tiles  = N_NODES / 16;           // 6250 exactly
    const int cBlk   = (tiles + 7) / 8;
    k_classify<<<cBlk, B, 0, stream>>>(buf2, W, bias, out);
}